// Lca_84018150245152
// MI455X (gfx1250) — compile-verified
//
#include <hip/hip_runtime.h>
#include <math.h>

typedef float v2f __attribute__((ext_vector_type(2)));
typedef float v4f __attribute__((ext_vector_type(4)));
typedef float v8f __attribute__((ext_vector_type(8)));

// Problem constants (from reference)
constexpr int Bn = 8;        // batch
constexpr int Cn = 64;       // dictionary atoms / channels
constexpr int Tn = 32000;    // input length
constexpr int Kn = 256;      // kernel length
constexpr int Sn = 7937;     // (T-K)/stride + 1, stride=4
constexpr int Qn = Tn / 4;   // 8000 output quads per batch
constexpr int STILES = (Sn + 15) / 16;   // 497
constexpr int WSTRIDE = 260;             // LDS row pad (260 mod 64 = 4 -> conflict-free)
constexpr float ETA = 0.1f;              // DT/TAU
constexpr float THRESH = 0.1f;

// ---- fp32 WMMA wrapper (V_WMMA_F32_16X16X4_F32) ----
__device__ __forceinline__ v8f wmma_f32_16x16x4(v2f a, v2f b, v8f c) {
#if __has_builtin(__builtin_amdgcn_wmma_f32_16x16x4_f32)
  return __builtin_amdgcn_wmma_f32_16x16x4_f32(false, a, false, b, (short)0, c,
                                               false, false);
#else
  c[0] += a.x * b.x + a.y * b.y;  // compile-safe fallback
  return c;
#endif
}

// ---- normalize dictionary atoms: wn[c,k] = w[c,k] / ||w[c,:]|| ----
__global__ __launch_bounds__(64) void lca_norm_kernel(
    const float* __restrict__ w, float* __restrict__ wn) {
  __shared__ float red[64];
  const int c = blockIdx.x;
  float s = 0.f;
  for (int k = threadIdx.x; k < Kn; k += 64) {
    float v = w[c * Kn + k];
    s += v * v;
  }
  red[threadIdx.x] = s;
  __syncthreads();
  for (int off = 32; off > 0; off >>= 1) {
    if (threadIdx.x < off) red[threadIdx.x] += red[threadIdx.x + off];
    __syncthreads();
  }
  const float inv = 1.0f / sqrtf(red[0]);
  for (int k = threadIdx.x; k < Kn; k += 64)
    wn[c * Kn + k] = w[c * Kn + k] * inv;
}

// ---- strided conv1d as WMMA GEMM: y[b,s,c] = sum_k xin[b,4s+k]*wn[c,k]
// mode 0: y -> drive
// mode 1: fused LCA update: u += eta*(drive - u - y + a); a = hardshrink(u);
//         active count -> loss (0.5*thresh^2/B per active)
__global__ __launch_bounds__(128) void lca_conv_kernel(
    const float* __restrict__ xin,    // [B][T]
    const float* __restrict__ wn,     // [64][256]
    float* __restrict__ drive,        // [B][S][64]
    float* __restrict__ u,            // [B][S][64]
    float* __restrict__ a,            // [B][S][64]
    float* __restrict__ loss, int mode) {
  __shared__ float w_lds[Cn * WSTRIDE];
  __shared__ float x_lds[320];
  __shared__ float red[128];

  const int tid = threadIdx.x;
  const int lane = tid & 31;
  const int wave = tid >> 5;
  const int b = blockIdx.y;

  for (int i = tid; i < Cn * Kn; i += 128)
    w_lds[(i >> 8) * WSTRIDE + (i & 255)] = wn[i];

  const int n = lane & 15;     // A row / B,C column within tile
  const int half = lane >> 4;  // K-half selector
  const int c0 = wave * 16;    // channel tile
  const float* wb = &w_lds[(c0 + n) * WSTRIDE + 2 * half];

  float cnt = 0.f;

  for (int st = blockIdx.x; st < STILES; st += gridDim.x) {
    const int s0 = st * 16;
    const int base = s0 * 4;
    __syncthreads();
    for (int i = tid; i < 316; i += 128) {
      const int idx = base + i;
      x_lds[i] = (idx < Tn) ? xin[(size_t)b * Tn + idx] : 0.f;
    }
    __syncthreads();

    const float* xa = &x_lds[4 * n + 2 * half];
    v8f acc = {};
#pragma unroll 8
    for (int k0 = 0; k0 < Kn; k0 += 4) {
      v2f af;
      af.x = xa[k0];
      af.y = xa[k0 + 1];
      v2f bf;
      bf.x = wb[k0];
      bf.y = wb[k0 + 1];
      acc = wmma_f32_16x16x4(af, bf, acc);
    }

#pragma unroll
    for (int v = 0; v < 8; ++v) {
      const int s = s0 + v + 8 * half;
      if (s < Sn) {
        const size_t idx = ((size_t)b * Sn + s) * 64 + c0 + n;
        if (mode == 0) {
          drive[idx] = acc[v];
        } else {
          float uu = u[idx];
          const float av = a[idx];
          uu += ETA * (drive[idx] - uu - acc[v] + av);
          const bool active = fabsf(uu) > THRESH;
          u[idx] = uu;
          a[idx] = active ? uu : 0.f;
          cnt += active ? 1.f : 0.f;
        }
      }
    }
  }

  if (mode == 1) {
    red[tid] = cnt;
    __syncthreads();
    for (int off = 64; off > 0; off >>= 1) {
      if (tid < off) red[tid] += red[tid + off];
      __syncthreads();
    }
    if (tid == 0)
      atomicAdd(loss, red[0] * (0.5f * THRESH * THRESH / (float)Bn));
  }
}

// ---- conv_transpose pre-pass as WMMA GEMM: P[b,s,j] = sum_c a[b,s,c]*wn[c,j]
__global__ __launch_bounds__(128) void lca_convt_kernel(
    const float* __restrict__ a,   // [B][S][64]
    const float* __restrict__ wn,  // [64][256]
    float* __restrict__ P) {       // [B][S][256]
  __shared__ float w_lds[Cn * WSTRIDE];
  const int tid = threadIdx.x;
  const int lane = tid & 31;
  const int wave = tid >> 5;
  const int b = blockIdx.y;

  for (int i = tid; i < Cn * Kn; i += 128)
    w_lds[(i >> 8) * WSTRIDE + (i & 255)] = wn[i];
  __syncthreads();

  const int n = lane & 15;
  const int half = lane >> 4;

  for (int st = blockIdx.x; st < STILES; st += gridDim.x) {
    const int s0 = st * 16;
    const int sA = s0 + n;

    // Preload all A fragments for this s-tile (K=64 channels -> 16 steps),
    // reused across the wave's 4 j-tiles.
    v2f afrag[16];
    const float* abase =
        &a[((size_t)b * Sn + (sA < Sn ? sA : 0)) * 64 + 2 * half];
#pragma unroll
    for (int k = 0; k < 16; ++k) {
      if (sA < Sn) {
        afrag[k].x = abase[4 * k];
        afrag[k].y = abase[4 * k + 1];
      } else {
        afrag[k].x = 0.f;
        afrag[k].y = 0.f;
      }
    }

#pragma unroll
    for (int t = 0; t < 4; ++t) {
      const int j0 = (wave * 4 + t) * 16;
      const float* wbp = &w_lds[(2 * half) * WSTRIDE + j0 + n];
      v8f acc = {};
#pragma unroll 8
      for (int k = 0; k < 16; ++k) {
        v2f bf;
        bf.x = wbp[(4 * k) * WSTRIDE];
        bf.y = wbp[(4 * k + 1) * WSTRIDE];
        acc = wmma_f32_16x16x4(afrag[k], bf, acc);
      }
#pragma unroll
      for (int v = 0; v < 8; ++v) {
        const int s = s0 + v + 8 * half;
        if (s < Sn) P[((size_t)b * Sn + s) * 256 + j0 + n] = acc[v];
      }
    }
  }
}

// ---- overlap-add + reconstruction loss (vectorized b128):
// one thread per output quad q: r[b,4q+m] = sum_d P[b,q-d,4d+m], m=0..3
// loss += 0.5*sum((r-x)^2)/B
__global__ __launch_bounds__(256) void lca_oa_loss_kernel(
    const float* __restrict__ P,  // [B][S][256]
    const float* __restrict__ x,  // [B][T]
    float* __restrict__ r,        // [B][T]
    float* __restrict__ loss) {
  __shared__ float red[256];
  const int b = blockIdx.y;
  const float* Pb = P + (size_t)b * Sn * 256;
  float local = 0.f;
  for (int q = blockIdx.x * 256 + threadIdx.x; q < Qn; q += gridDim.x * 256) {
    const int dlo = (q - (Sn - 1) > 0) ? (q - (Sn - 1)) : 0;
    const int dhi = (q < 63) ? q : 63;
    v4f sum = {0.f, 0.f, 0.f, 0.f};
    for (int d = dlo; d <= dhi; ++d) {
      // 16B-aligned contiguous quad: P[q-d][4d + 0..3]
      const v4f p = *(const v4f*)&Pb[(size_t)(q - d) * 256 + 4 * d];
      sum += p;
    }
    *(v4f*)&r[(size_t)b * Tn + 4 * q] = sum;
    const v4f xv = *(const v4f*)&x[(size_t)b * Tn + 4 * q];
    const v4f diff = sum - xv;
    local += diff.x * diff.x + diff.y * diff.y + diff.z * diff.z +
             diff.w * diff.w;
  }
  red[threadIdx.x] = local;
  __syncthreads();
  for (int off = 128; off > 0; off >>= 1) {
    if (threadIdx.x < off) red[threadIdx.x] += red[threadIdx.x + off];
    __syncthreads();
  }
  if (threadIdx.x == 0) atomicAdd(loss, red[0] * (0.5f / (float)Bn));
}

extern "C" void kernel_launch(void* const* d_in, const int* in_sizes, int n_in,
                              void* d_out, int out_size, void* d_ws,
                              size_t ws_size, hipStream_t stream) {
  const float* x = (const float*)d_in[0];  // [8,1,32000]
  const float* w = (const float*)d_in[1];  // [64,1,256]
  float* out = (float*)d_out;              // [1] loss

  // Workspace layout (floats): ~115 MB total, L2-resident working set
  float* ws = (float*)d_ws;
  size_t off = 0;
  float* wn = ws + off;    off += (size_t)Cn * Kn;        //   64 KB
  float* drive = ws + off; off += (size_t)Bn * Sn * 64;   // 16.3 MB
  float* u = ws + off;     off += (size_t)Bn * Sn * 64;   // 16.3 MB
  float* a = ws + off;     off += (size_t)Bn * Sn * 64;   // 16.3 MB
  float* r = ws + off;     off += (size_t)Bn * Tn;        //  1.0 MB
  float* P = ws + off;     off += (size_t)Bn * Sn * 256;  // 65.0 MB

  // Deterministic init each call (harness does not re-poison between replays)
  hipMemsetAsync(out, 0, sizeof(float), stream);
  hipMemsetAsync(u, 0, (size_t)Bn * Sn * 64 * sizeof(float), stream);
  hipMemsetAsync(a, 0, (size_t)Bn * Sn * 64 * sizeof(float), stream);
  hipMemsetAsync(r, 0, (size_t)Bn * Tn * sizeof(float), stream);

  lca_norm_kernel<<<dim3(Cn), dim3(64), 0, stream>>>(w, wn);

  const dim3 cgrid(256, Bn), cblock(128);
  // Feedforward drive (computed once; constant over iterations)
  lca_conv_kernel<<<cgrid, cblock, 0, stream>>>(x, wn, drive, u, a, out, 0);

  for (int it = 0; it < 5; ++it) {
    // inhib = conv(r, wn) fused with membrane update + sparsity cost
    // (r==0 on first iteration -> inhib==0, matching reference)
    lca_conv_kernel<<<cgrid, cblock, 0, stream>>>(r, wn, drive, u, a, out, 1);
    // P[s,j] = sum_c a[s,c] * wn[c,j]  (convT pre-pass GEMM)
    lca_convt_kernel<<<cgrid, cblock, 0, stream>>>(a, wn, P);
    // r = overlap-add(P); loss += 0.5*||r - x||^2 / B
    // (this r feeds the next iteration's inhib conv)
    lca_oa_loss_kernel<<<dim3(32, Bn), dim3(256), 0, stream>>>(P, x, r, out);
  }
}